// GIKT_15152644620314
// MI455X (gfx1250) — compile-verified
//
#include <hip/hip_runtime.h>
#include <hip/hip_bf16.h>

// GIKT forward for MI455X (gfx1250, wave32, WMMA f32 16x16x4).
// Phase kernels:
//   k_pre1 : per-t graph aggregation + GRU1 input transform GI1[t] (WMMA)
//   k_pre2 : qw[t,b,q] = tanh(qc @ W_key^T + b_key) . w_q        (WMMA)
//   k_pre3 : top-k recall indices from embedding dot scores
//   k_seq  : persistent single-workgroup 63-step recurrence (WMMA)

#define NQc 100000
#define NCc 2000
#define DD  128
#define BB  128
#define SS  64
#define TSTEPS 63
#define RANKK 10

typedef float v2f __attribute__((ext_vector_type(2)));
typedef float v8f __attribute__((ext_vector_type(8)));

__device__ __forceinline__ v8f wmma4(v2f a, v2f b, v8f c) {
  // D = A(16x4) * B(4x16) + C(16x16), all f32. 8-arg VOP3P form.
  return __builtin_amdgcn_wmma_f32_16x16x4_f32(false, a, false, b, (short)0, c,
                                               false, false);
}
__device__ __forceinline__ float sigmoidf_(float x) {
  return 1.0f / (1.0f + expf(-x));
}

// Out(128x128, LDS) = tanh((A [+A2]) @ W^T + bias)    (MODE 0)
// Out            += 0.25f * tanh(A @ W^T + bias)      (MODE 1)
// A, A2: 128x128 LDS row-major. W: (>=128, 128) global row-major (N,K).
// Force-inlined template so address spaces stay concrete:
//   A/A2/Out -> LDS (ds_load_b64), W/bias -> global (global_load).
template <int MODE, bool HASA2>
__device__ __forceinline__ void gemm_dd(const float* __restrict__ A,
                                        const float* __restrict__ A2,
                                        const float* __restrict__ W,
                                        const float* __restrict__ bias,
                                        float* __restrict__ Out) {
  const int wave = threadIdx.x >> 5;
  const int lane = threadIdx.x & 31;
  const int kh = (lane >> 4) << 1;
  for (int tile = wave; tile < 64; tile += 32) {
    const int mt = tile >> 3, nt = tile & 7;
    const int arow = ((mt << 4) + (lane & 15)) * DD + kh;
    const int ncol = (nt << 4) + (lane & 15);
    const float* wr = W + (size_t)ncol * DD + kh;
    v8f acc = {};
#pragma unroll
    for (int k = 0; k < DD; k += 4) {
      v2f av = *(const v2f*)(A + arow + k);
      if constexpr (HASA2) av += *(const v2f*)(A2 + arow + k);
      acc = wmma4(av, *(const v2f*)(wr + k), acc);
    }
    const float bb = bias[ncol];
    const int mbase = (mt << 4) + ((lane >> 4) << 3);
#pragma unroll
    for (int v = 0; v < 8; ++v) {
      const float r = tanhf(acc[v] + bb);
      if (MODE) Out[(mbase + v) * DD + ncol] += 0.25f * r;
      else      Out[(mbase + v) * DD + ncol] = r;
    }
  }
}

// ---------------- Precompute 1: aggregation + GI1 --------------------------
__global__ __launch_bounds__(1024) void k_pre1(
    const float* __restrict__ EQ, const float* __restrict__ EC,
    const float* __restrict__ ECORR,
    const float* __restrict__ Wih1, const float* __restrict__ bih1,
    const float* __restrict__ Wagg, const float* __restrict__ bagg,
    const float* __restrict__ WL, const float* __restrict__ bL,
    const int* __restrict__ qnbr, const int* __restrict__ cnbr,
    const int* __restrict__ qseq, const int* __restrict__ corr,
    const int* __restrict__ mseq, float* __restrict__ GI1) {
  extern __shared__ float sm[];
  float* P0  = sm;              // stage-A input, later emb_q (XQ)
  float* PU0 = sm + 16384;      // u0
  float* PV  = sm + 2 * 16384;  // vin / u1
  float* PMV = sm + 3 * 16384;  // mean(v) accumulator
  float* PXR = sm + 4 * 16384;  // emb_r (XR)
  const int t = blockIdx.x;

  for (int i = threadIdx.x; i < BB * DD; i += blockDim.x) {
    const int b = i >> 7, d = i & 127;
    const int q = qseq[b * SS + t];
    const int* nb = qnbr + q * 4;
    P0[i] = EQ[(size_t)q * DD + d] +
            0.25f * (EC[(size_t)nb[0] * DD + d] + EC[(size_t)nb[1] * DD + d] +
                     EC[(size_t)nb[2] * DD + d] + EC[(size_t)nb[3] * DD + d]);
    PMV[i] = 0.0f;
    PXR[i] = ECORR[(size_t)corr[b * SS + t] * DD + d];
  }
  __syncthreads();
  gemm_dd<0, false>(P0, nullptr, Wagg, bagg, PU0);  // u0
  __syncthreads();
  for (int j = 0; j < 4; ++j) {
    for (int i = threadIdx.x; i < BB * DD; i += blockDim.x) {
      const int b = i >> 7, d = i & 127;
      const int q = qseq[b * SS + t];
      const int c = qnbr[q * 4 + j];
      const int* cn = cnbr + c * 4;
      PV[i] = EC[(size_t)c * DD + d] +
              0.25f * (EQ[(size_t)cn[0] * DD + d] + EQ[(size_t)cn[1] * DD + d] +
                       EQ[(size_t)cn[2] * DD + d] + EQ[(size_t)cn[3] * DD + d]);
    }
    __syncthreads();
    gemm_dd<1, false>(PV, nullptr, Wagg + DD * DD, bagg + DD, PMV);  // +=mean(v)
    __syncthreads();
  }
  gemm_dd<0, true>(PMV, PU0, Wagg, bagg, PV);  // u1 = tanh((mv+u0)Wg0^T+bg0)
  __syncthreads();
  {  // emb_q = mask ? tanh(u1 WL^T + bL) : EQ[q_t]
    const int wave = threadIdx.x >> 5, lane = threadIdx.x & 31;
    const int kh = (lane >> 4) << 1;
    for (int tile = wave; tile < 64; tile += 32) {
      const int mt = tile >> 3, nt = tile & 7;
      const int arow = ((mt << 4) + (lane & 15)) * DD + kh;
      const int ncol = (nt << 4) + (lane & 15);
      const float* wr = WL + (size_t)ncol * DD + kh;
      v8f acc = {};
#pragma unroll
      for (int k = 0; k < DD; k += 4)
        acc = wmma4(*(const v2f*)(PV + arow + k), *(const v2f*)(wr + k), acc);
      const float bb = bL[ncol];
      const int mbase = (mt << 4) + ((lane >> 4) << 3);
#pragma unroll
      for (int v = 0; v < 8; ++v) {
        const int b = mbase + v;
        const float aggv = tanhf(acc[v] + bb);
        const int q = qseq[b * SS + t];
        P0[b * DD + ncol] =
            (mseq[b * SS + t] != 0) ? aggv : EQ[(size_t)q * DD + ncol];
      }
    }
  }
  __syncthreads();
  {  // GI1[t] = [emb_q | emb_r] @ Wih1^T + bih1   (M=128,N=384,K=256)
    const int wave = threadIdx.x >> 5, lane = threadIdx.x & 31;
    const int kh = (lane >> 4) << 1;
    for (int tile = wave; tile < 8 * 24; tile += 32) {
      const int mt = tile / 24, nt = tile % 24;
      const int arow = ((mt << 4) + (lane & 15)) * DD + kh;
      const int ncol = (nt << 4) + (lane & 15);
      const float* wr = Wih1 + (size_t)ncol * (2 * DD) + kh;
      v8f acc = {};
#pragma unroll
      for (int k = 0; k < DD; k += 4)
        acc = wmma4(*(const v2f*)(P0 + arow + k), *(const v2f*)(wr + k), acc);
#pragma unroll
      for (int k = 0; k < DD; k += 4)
        acc = wmma4(*(const v2f*)(PXR + arow + k), *(const v2f*)(wr + DD + k), acc);
      const float bb = bih1[ncol];
      const int mbase = (mt << 4) + ((lane >> 4) << 3);
#pragma unroll
      for (int v = 0; v < 8; ++v)
        GI1[((size_t)t * BB + (mbase + v)) * 384 + ncol] = acc[v] + bb;
    }
  }
}

// ---------------- Precompute 2: qw = tanh(qc W_key^T + b_key) . w_q --------
__global__ __launch_bounds__(128) void k_pre2(
    const float* __restrict__ EQ, const float* __restrict__ EC,
    const float* __restrict__ Wkey, const float* __restrict__ bkey,
    const float* __restrict__ wq, const int* __restrict__ q2c,
    const int* __restrict__ qseq, float* __restrict__ QW) {
  __shared__ float At[16 * DD];
  __shared__ float qrow[16];
  const int mt = blockIdx.x;  // rows r = mt*16 .. +15, r = (t*128+b)*5+q
  for (int i = threadIdx.x; i < 16 * DD; i += blockDim.x) {
    const int rr = i >> 7, d = i & 127;
    const int r = mt * 16 + rr;
    const int t = r / 640, rem = r % 640;
    const int b = rem / 5, q = rem % 5;
    const int qn = qseq[b * SS + t + 1];
    const float* row = (q == 0) ? (EQ + (size_t)qn * DD)
                                : (EC + (size_t)q2c[qn * 4 + (q - 1)] * DD);
    At[i] = row[d];
  }
  if (threadIdx.x < 16) qrow[threadIdx.x] = 0.0f;
  __syncthreads();
  const int wave = threadIdx.x >> 5, lane = threadIdx.x & 31;
  const int kh = (lane >> 4) << 1;
  for (int nt = wave; nt < 8; nt += 4) {
    const int ncol = (nt << 4) + (lane & 15);
    const float* wr = Wkey + (size_t)ncol * DD + kh;
    const float* a = At + (lane & 15) * DD + kh;
    v8f acc = {};
#pragma unroll
    for (int k = 0; k < DD; k += 4)
      acc = wmma4(*(const v2f*)(a + k), *(const v2f*)(wr + k), acc);
    const float bb = bkey[ncol], w = wq[ncol];
    const int mbase = (lane >> 4) << 3;
#pragma unroll
    for (int v = 0; v < 8; ++v)
      atomicAdd(&qrow[mbase + v], tanhf(acc[v] + bb) * w);
  }
  __syncthreads();
  if (threadIdx.x < 16) QW[(size_t)mt * 16 + threadIdx.x] = qrow[threadIdx.x];
}

// ---------------- Precompute 3: top-k history indices ----------------------
__global__ __launch_bounds__(64) void k_pre3(const float* __restrict__ EQ,
                                             const int* __restrict__ qseq,
                                             int* __restrict__ TOPK) {
  __shared__ float sc[64];
  const int bidx = blockIdx.x;
  const int b = bidx / 52;
  const int t = 11 + (bidx % 52);
  const int qn = qseq[b * SS + t + 1];
  const float* Q = EQ + (size_t)qn * DD;
  for (int j = threadIdx.x; j < t; j += 64) {
    const float* K = EQ + (size_t)qseq[b * SS + j] * DD;
    float a = 0.0f;
    for (int d = 0; d < DD; ++d) a += Q[d] * K[d];
    sc[j] = a;
  }
  __syncthreads();
  if (threadIdx.x == 0) {  // stable descending top-10 (ties -> lower index)
    unsigned long long used = 0ull;
    for (int i = 0; i < RANKK; ++i) {
      float best = -3.4e38f; int bj = 0;
      for (int j = 0; j < t; ++j)
        if (!((used >> j) & 1ull) && sc[j] > best) { best = sc[j]; bj = j; }
      used |= 1ull << bj;
      TOPK[((size_t)b * TSTEPS + t) * RANKK + i] = bj;
    }
  }
}

// ---------------- Sequential recurrence (persistent workgroup) -------------
__global__ __launch_bounds__(1024) void k_seq(
    const float* __restrict__ EQ, const float* __restrict__ EC,
    const float* __restrict__ Whh1, const float* __restrict__ bhh1,
    const float* __restrict__ Wih2, const float* __restrict__ bih2,
    const float* __restrict__ Whh2, const float* __restrict__ bhh2,
    const float* __restrict__ Wquery, const float* __restrict__ bquery,
    const float* __restrict__ wk, const float* __restrict__ bw_p,
    const float* __restrict__ h1i, const float* __restrict__ h2i,
    const int* __restrict__ q2c, const int* __restrict__ qseq,
    const float* __restrict__ GI1, const float* __restrict__ QW,
    const int* __restrict__ TOPK, float* __restrict__ HIST,
    float* __restrict__ KW, float* __restrict__ y) {
  extern __shared__ float sm[];
  float* H1 = sm;                  // double buffer: 2 * 16384
  float* H2 = sm + 2 * 16384;
  float* G2 = sm + 3 * 16384;
  float* OG = sm + 4 * 16384;      // 7040
  float* kwc = OG + 7040;          // 128
  float* kwz = kwc + 128;          // 1
  const int tid = threadIdx.x;
  const int wave = tid >> 5, lane = tid & 31;
  const int kh = (lane >> 4) << 1;
  const float bw = bw_p[0];

  for (int i = tid; i < BB * DD; i += blockDim.x) { H1[i] = h1i[i]; H2[i] = h2i[i]; }
  for (int i = tid; i < BB * SS; i += blockDim.x) y[i] = 0.0f;
  for (int i = tid; i < BB * SS * DD; i += blockDim.x) HIST[i] = 0.0f;
  if (tid == 0) kwz[0] = 0.0f;
  __syncthreads();
  if (tid < DD) atomicAdd(kwz, tanhf(bquery[tid]) * wk[tid]);  // kw of zero state
  __syncthreads();
  const float kz = kwz[0];
  for (int i = tid; i < BB * SS; i += blockDim.x) KW[i] = kz;
  __syncthreads();

  for (int t = 0; t < TSTEPS; ++t) {
    float* h1c = H1 + ((t & 1) ? 16384 : 0);
    float* h1n = H1 + ((t & 1) ? 0 : 16384);

    // ---- GRU1: gh1 = h1 @ Whh1^T (+bhh1), fused gate epilogue ----
    for (int tile = wave; tile < 64; tile += 32) {
      const int mt = tile >> 3, nt = tile & 7;
      const int arow = ((mt << 4) + (lane & 15)) * DD + kh;
      const int ncol = (nt << 4) + (lane & 15);
      const float* wr = Whh1 + (size_t)(ncol         ) * DD + kh;
      const float* wz = Whh1 + (size_t)(ncol + DD    ) * DD + kh;
      const float* wn = Whh1 + (size_t)(ncol + 2 * DD) * DD + kh;
      v8f ar = {}, az = {}, an = {};
#pragma unroll
      for (int k = 0; k < DD; k += 4) {
        const v2f av = *(const v2f*)(h1c + arow + k);
        ar = wmma4(av, *(const v2f*)(wr + k), ar);
        az = wmma4(av, *(const v2f*)(wz + k), az);
        an = wmma4(av, *(const v2f*)(wn + k), an);
      }
      const float br = bhh1[ncol], bz = bhh1[ncol + DD], bn = bhh1[ncol + 2 * DD];
      const int mbase = (mt << 4) + ((lane >> 4) << 3);
#pragma unroll
      for (int v = 0; v < 8; ++v) {
        const int m = mbase + v;
        const float* gi = GI1 + ((size_t)t * BB + m) * 384 + ncol;
        const float r = sigmoidf_(gi[0] + ar[v] + br);
        const float z = sigmoidf_(gi[DD] + az[v] + bz);
        const float nn = tanhf(gi[2 * DD] + r * (an[v] + bn));
        h1n[m * DD + ncol] = (1.0f - z) * nn + z * h1c[m * DD + ncol];
      }
    }
    __syncthreads();

    // ---- GRU2: gi2 = h1' @ Wih2^T, gh2 = h2 @ Whh2^T, fused epilogue ----
    for (int tile = wave; tile < 64; tile += 32) {
      const int mt = tile >> 3, nt = tile & 7;
      const int arow = ((mt << 4) + (lane & 15)) * DD + kh;
      const int ncol = (nt << 4) + (lane & 15);
      const float* ir_ = Wih2 + (size_t)(ncol         ) * DD + kh;
      const float* iz_ = Wih2 + (size_t)(ncol + DD    ) * DD + kh;
      const float* in_ = Wih2 + (size_t)(ncol + 2 * DD) * DD + kh;
      const float* hr_ = Whh2 + (size_t)(ncol         ) * DD + kh;
      const float* hz_ = Whh2 + (size_t)(ncol + DD    ) * DD + kh;
      const float* hn_ = Whh2 + (size_t)(ncol + 2 * DD) * DD + kh;
      v8f gir = {}, giz = {}, gin = {}, ghr = {}, ghz = {}, ghn = {};
#pragma unroll
      for (int k = 0; k < DD; k += 4) {
        const v2f a1 = *(const v2f*)(h1n + arow + k);
        const v2f a2 = *(const v2f*)(H2 + arow + k);
        gir = wmma4(a1, *(const v2f*)(ir_ + k), gir);
        giz = wmma4(a1, *(const v2f*)(iz_ + k), giz);
        gin = wmma4(a1, *(const v2f*)(in_ + k), gin);
        ghr = wmma4(a2, *(const v2f*)(hr_ + k), ghr);
        ghz = wmma4(a2, *(const v2f*)(hz_ + k), ghz);
        ghn = wmma4(a2, *(const v2f*)(hn_ + k), ghn);
      }
      const float bir = bih2[ncol], biz = bih2[ncol + DD], bin = bih2[ncol + 2 * DD];
      const float bhr = bhh2[ncol], bhz = bhh2[ncol + DD], bhn = bhh2[ncol + 2 * DD];
      const int mbase = (mt << 4) + ((lane >> 4) << 3);
#pragma unroll
      for (int v = 0; v < 8; ++v) {
        const int m = mbase + v;
        const float r = sigmoidf_(gir[v] + bir + ghr[v] + bhr);
        const float z = sigmoidf_(giz[v] + biz + ghz[v] + bhz);
        const float nn = tanhf(gin[v] + bin + r * (ghn[v] + bhn));
        G2[m * DD + ncol] = (1.0f - z) * nn + z * H2[m * DD + ncol];
      }
    }
    if (tid < BB) kwc[tid] = 0.0f;
    __syncthreads();

    // ---- kw_cur = tanh(g2 @ Wquery^T + bquery) . w_k  (LDS atomic reduce) ----
    for (int tile = wave; tile < 64; tile += 32) {
      const int mt = tile >> 3, nt = tile & 7;
      const int arow = ((mt << 4) + (lane & 15)) * DD + kh;
      const int ncol = (nt << 4) + (lane & 15);
      const float* wr = Wquery + (size_t)ncol * DD + kh;
      v8f acc = {};
#pragma unroll
      for (int k = 0; k < DD; k += 4)
        acc = wmma4(*(const v2f*)(G2 + arow + k), *(const v2f*)(wr + k), acc);
      const float bb = bquery[ncol], w = wk[ncol];
      const int mbase = (mt << 4) + ((lane >> 4) << 3);
#pragma unroll
      for (int v = 0; v < 8; ++v)
        atomicAdd(&kwc[mbase + v], tanhf(acc[v] + bb) * w);
    }
    __syncthreads();

    // ---- og[b,q,s] = qc[b,q] . states[b,s] ----
    const int L = (t <= RANKK) ? (t + 1) : (RANKK + 1);
    for (int idx = tid; idx < BB * 5 * L; idx += blockDim.x) {
      const int b = idx / (5 * L);
      const int rem = idx - b * 5 * L;
      const int q = rem / L, s = rem - q * L;
      const int qn = qseq[b * SS + t + 1];
      const float* qc = (q == 0) ? (EQ + (size_t)qn * DD)
                                 : (EC + (size_t)q2c[qn * 4 + (q - 1)] * DD);
      const float* st;
      if (s == 0) st = G2 + b * DD;
      else {
        const int j = (t <= RANKK)
                          ? (s - 1)
                          : TOPK[((size_t)b * TSTEPS + t) * RANKK + (s - 1)];
        st = HIST + ((size_t)b * SS + j) * DD;
      }
      float a = 0.0f;
      for (int d = 0; d < DD; ++d) a += qc[d] * st[d];
      OG[idx] = a;
    }
    __syncthreads();

    // ---- softmax attention + sigmoid output (one thread per b) ----
    if (tid < BB) {
      const int b = tid;
      float kwv[RANKK + 1];
      kwv[0] = kwc[b];
      for (int s = 1; s < L; ++s) {
        const int j = (t <= RANKK)
                          ? (s - 1)
                          : TOPK[((size_t)b * TSTEPS + t) * RANKK + (s - 1)];
        kwv[s] = KW[b * SS + j];
      }
      float total = 0.0f;
      for (int q = 0; q < 5; ++q) {
        const float qv = QW[((size_t)t * BB + b) * 5 + q];
        float mx = -3.4e38f;
        for (int s = 0; s < L; ++s) {
          const float tm = qv + kwv[s] + bw;
          if (tm > mx) mx = tm;
        }
        float den = 0.0f, num = 0.0f;
        const float* ogr = OG + (b * 5 + q) * L;
        for (int s = 0; s < L; ++s) {
          const float e = expf(qv + kwv[s] + bw - mx);
          den += e; num += e * ogr[s];
        }
        total += num / den;
      }
      y[b * SS + ((t == 0) ? 0 : (t + 1))] = sigmoidf_(total);
    }
    __syncthreads();

    // ---- state update (skipped at t==0, matching reference `continue`) ----
    if (t >= 1) {
      for (int i = tid; i < BB * DD; i += blockDim.x) {
        const float g = G2[i];
        H2[i] = g;
        const int b = i >> 7, d = i & 127;
        HIST[((size_t)b * SS + t) * DD + d] = g;
      }
      if (tid < BB) KW[tid * SS + t] = kwc[tid];
    }
    __syncthreads();
  }
}

extern "C" void kernel_launch(void* const* d_in, const int* in_sizes, int n_in,
                              void* d_out, int out_size, void* d_ws,
                              size_t ws_size, hipStream_t stream) {
  (void)in_sizes; (void)n_in; (void)out_size; (void)ws_size;
  const float* EQ    = (const float*)d_in[0];
  const float* EC    = (const float*)d_in[1];
  const float* ECORR = (const float*)d_in[2];
  const float* Wih1  = (const float*)d_in[3];
  const float* Whh1  = (const float*)d_in[4];
  const float* bih1  = (const float*)d_in[5];
  const float* bhh1  = (const float*)d_in[6];
  const float* Wih2  = (const float*)d_in[7];
  const float* Whh2  = (const float*)d_in[8];
  const float* bih2  = (const float*)d_in[9];
  const float* bhh2  = (const float*)d_in[10];
  const float* Wagg  = (const float*)d_in[11];
  const float* bagg  = (const float*)d_in[12];
  const float* WL    = (const float*)d_in[13];
  const float* bL    = (const float*)d_in[14];
  const float* Wqry  = (const float*)d_in[15];
  const float* bqry  = (const float*)d_in[16];
  const float* Wkey  = (const float*)d_in[17];
  const float* bkey  = (const float*)d_in[18];
  const float* wq    = (const float*)d_in[19];
  const float* wk    = (const float*)d_in[20];
  const float* bw    = (const float*)d_in[21];
  const float* h1i   = (const float*)d_in[22];
  const float* h2i   = (const float*)d_in[23];
  const int* qnbr    = (const int*)d_in[24];
  const int* cnbr    = (const int*)d_in[25];
  const int* q2c     = (const int*)d_in[26];
  const int* qseq    = (const int*)d_in[27];
  const int* corr    = (const int*)d_in[28];
  const int* mseq    = (const int*)d_in[29];
  float* y = (float*)d_out;

  float* ws = (float*)d_ws;
  float* GI1  = ws;                                  // 63*128*384
  float* QW   = GI1 + (size_t)TSTEPS * BB * 384;     // 63*128*5
  int*   TOPK = (int*)(QW + (size_t)TSTEPS * BB * 5);// 128*63*10
  float* HIST = (float*)(TOPK + (size_t)BB * TSTEPS * RANKK);  // 128*64*128
  float* KW   = HIST + (size_t)BB * SS * DD;         // 128*64

  const size_t lds_pre1 = (size_t)5 * 16384 * sizeof(float);            // 320 KB
  const size_t lds_seq  = (size_t)(4 * 16384 + 7040 + 128 + 8) * sizeof(float);

  k_pre1<<<TSTEPS, 1024, lds_pre1, stream>>>(EQ, EC, ECORR, Wih1, bih1, Wagg,
                                             bagg, WL, bL, qnbr, cnbr, qseq,
                                             corr, mseq, GI1);
  k_pre2<<<(TSTEPS * BB * 5) / 16, 128, 0, stream>>>(EQ, EC, Wkey, bkey, wq,
                                                     q2c, qseq, QW);
  k_pre3<<<BB * 52, 64, 0, stream>>>(EQ, qseq, TOPK);
  k_seq<<<1, 1024, lds_seq, stream>>>(EQ, EC, Whh1, bhh1, Wih2, bih2, Whh2,
                                      bhh2, Wqry, bqry, wk, bw, h1i, h2i, q2c,
                                      qseq, GI1, QW, TOPK, HIST, KW, y);
}